// GCNModel_6210522710414
// MI455X (gfx1250) — compile-verified
//
#include <hip/hip_runtime.h>
#include <stdint.h>

typedef __attribute__((ext_vector_type(2)))  float  v2f;
typedef __attribute__((ext_vector_type(8)))  float  v8f;

#define BN_EPS 1e-5f

static inline unsigned cdiv(unsigned a, unsigned b) { return (a + b - 1) / b; }

// ---------------------------------------------------------------- degrees ---
__global__ void deg_init_kernel(float* __restrict__ deg, int n) {
    int i = blockIdx.x * blockDim.x + threadIdx.x;
    if (i < n) deg[i] = 1.0f;                       // self-loop contribution
}

__global__ void deg_accum_kernel(const int* __restrict__ dst, int E,
                                 float* __restrict__ deg) {
    int e = blockIdx.x * blockDim.x + threadIdx.x;
    if (e < E) atomicAdd(&deg[dst[e]], 1.0f);
}

__global__ void deg_rsqrt_kernel(float* __restrict__ deg, int n) {
    int i = blockIdx.x * blockDim.x + threadIdx.x;
    if (i < n) {
        float d = deg[i];
        deg[i] = d > 0.0f ? rsqrtf(d) : 0.0f;       // dinv
    }
}

// ------------------------------------------------------------------- GEMM ---
// C[M,N] = A[M,K] @ B[K,N], fp32 row-major, K % 32 == 0.
// Block: 256 threads = 8 waves. Block tile 128x64 (wave = 16 rows x 64 cols,
// 4 accumulators sharing one A fragment). K chunked by 32 through a
// double-buffered LDS B tile; next chunk's B global loads overlap the 32
// WMMAs of the current chunk. One barrier per chunk.
//
// LDS layout: B chunk pre-swizzled into fragment order. Segment (s,h) holds
// row-pair (k+4s+2h, k+4s+2h+1) for 64 columns as float2; segment stride
// 160 dwords (== 32 mod 64) -> half-wave b64 reads hit disjoint bank halves.
#define KCHUNK 32
#define NSEG   16            // (KCHUNK/4) * 2 halves
#define SEGSTR 160           // dwords per segment
#define BUFSZ  (NSEG * SEGSTR)

__global__ __launch_bounds__(256)
void gemm_wmma_f32(const float* __restrict__ A, const float* __restrict__ B,
                   float* __restrict__ C, int M, int N, int K) {
    __shared__ float bs[2 * BUFSZ];

    const int tid  = threadIdx.x;
    const int lane = tid & 31;
    const int wave = tid >> 5;              // 0..7
    const int m0   = blockIdx.x * 128 + wave * 16;
    const int n0   = blockIdx.y * 64;
    const int half = lane >> 4;             // 0 or 1
    const int l16  = lane & 15;

    v8f acc0 = {}, acc1 = {}, acc2 = {}, acc3 = {};

    // A fragment: lane holds row m0+l16, K-pair selected by half. Clamp row so
    // loads are always in-bounds (garbage rows are dropped by the store guard).
    const int row = min(m0 + l16, M - 1);
    const float* __restrict__ arow = A + (size_t)row * K + half * 2;

    // Per-thread B staging assignment: 4 (segment, column) pairs.
    int segi[4], ldsoff[4];
    size_t goff[4];
#pragma unroll
    for (int i = 0; i < 4; ++i) {
        const int p  = tid + i * 256;               // 0..1023
        const int c  = p & 63;                      // column within tile
        segi[i]      = p >> 6;                      // 0..15
        const int rr = (segi[i] >> 1) * 4 + (segi[i] & 1) * 2;  // K row in chunk
        const int gc = min(n0 + c, N - 1);          // clamped global column
        goff[i]      = (size_t)rr * N + gc;
        ldsoff[i]    = segi[i] * SEGSTR + c * 2;
    }

    // Prologue: fetch + stage chunk 0.
    float2 breg[4];
#pragma unroll
    for (int i = 0; i < 4; ++i) {
        breg[i].x = B[goff[i]];
        breg[i].y = B[goff[i] + N];
    }
#pragma unroll
    for (int i = 0; i < 4; ++i)
        *reinterpret_cast<float2*>(&bs[ldsoff[i]]) = breg[i];
    __syncthreads();

    const int nchunks = K / KCHUNK;
    for (int c = 0; c < nchunks; ++c) {
        const int    kcur = c * KCHUNK;
        const float* cur  = bs + (c & 1) * BUFSZ;
        float*       nxt  = bs + ((c + 1) & 1) * BUFSZ;
        const bool   more = (c + 1) < nchunks;

        // Prefetch next chunk's B into registers (overlaps compute below).
        if (more) {
            const float* bn = B + (size_t)(kcur + KCHUNK) * N;
#pragma unroll
            for (int i = 0; i < 4; ++i) {
                breg[i].x = bn[goff[i]];
                breg[i].y = bn[goff[i] + N];
            }
        }

        // Preload all 8 A fragments for this chunk (one clause, one wait).
        v2f a[8];
#pragma unroll
        for (int s = 0; s < 8; ++s)
            a[s] = *reinterpret_cast<const v2f*>(arow + kcur + 4 * s);

        // 8 K-steps of 4: 32 WMMAs.
#pragma unroll
        for (int s = 0; s < 8; ++s) {
            const float* bseg = cur + (2 * s + half) * SEGSTR + l16 * 2;
            v2f b0 = *reinterpret_cast<const v2f*>(bseg + 0 * 32);
            v2f b1 = *reinterpret_cast<const v2f*>(bseg + 1 * 32);
            v2f b2 = *reinterpret_cast<const v2f*>(bseg + 2 * 32);
            v2f b3 = *reinterpret_cast<const v2f*>(bseg + 3 * 32);
            acc0 = __builtin_amdgcn_wmma_f32_16x16x4_f32(false, a[s], false, b0, (short)0, acc0, false, false);
            acc1 = __builtin_amdgcn_wmma_f32_16x16x4_f32(false, a[s], false, b1, (short)0, acc1, false, false);
            acc2 = __builtin_amdgcn_wmma_f32_16x16x4_f32(false, a[s], false, b2, (short)0, acc2, false, false);
            acc3 = __builtin_amdgcn_wmma_f32_16x16x4_f32(false, a[s], false, b3, (short)0, acc3, false, false);
        }

        // Stage next chunk into the other LDS buffer.
        if (more) {
#pragma unroll
            for (int i = 0; i < 4; ++i)
                *reinterpret_cast<float2*>(&nxt[ldsoff[i]]) = breg[i];
        }
        __syncthreads();
    }

    // D layout: VGPR r -> row = m0 + 8*half + r, column = lane%16 (+16*j)
    const v8f accs[4] = {acc0, acc1, acc2, acc3};
#pragma unroll
    for (int j = 0; j < 4; ++j) {
        const int col = n0 + j * 16 + l16;
        if (col >= N) continue;
#pragma unroll
        for (int r = 0; r < 8; ++r) {
            const int rr = m0 + half * 8 + r;
            if (rr < M) C[(size_t)rr * N + col] = accs[j][r];
        }
    }
}

static inline void launch_gemm(const float* A, const float* B, float* C,
                               int M, int N, int K, hipStream_t stream) {
    dim3 grid(cdiv(M, 128), cdiv(N, 64));
    gemm_wmma_f32<<<grid, 256, 0, stream>>>(A, B, C, M, N, K);
}

// ------------------------------------------------------------ aggregation ---
// out[node][c] = bias[c] + h[node][c] * dinv[node]^2   (self-loop folded in)
// grid: (cdiv(C,256), n)
__global__ void agg_init_kernel(const float* __restrict__ h,
                                const float* __restrict__ dinv,
                                const float* __restrict__ bias,
                                float* __restrict__ out, int C) {
    const int node = blockIdx.y;
    const int c    = blockIdx.x * blockDim.x + threadIdx.x;
    if (c < C) {
        const size_t i = (size_t)node * C + c;
        const float  w = dinv[node];
        out[i] = fmaf(h[i], w * w, bias[c]);
    }
}

// one block per edge; float4 gathers from h[src], atomic adds into out[dst]
__global__ __launch_bounds__(128)
void agg_edges_kernel(const float* __restrict__ h,
                      const float* __restrict__ dinv,
                      const int* __restrict__ src,
                      const int* __restrict__ dst,
                      float* __restrict__ out, int C) {
    const int e = blockIdx.x;
    const int s = src[e];
    const int d = dst[e];
    const float w = dinv[s] * dinv[d];
    const float* __restrict__ hr = h + (size_t)s * C;
    float* __restrict__ orow     = out + (size_t)d * C;
    for (int c = threadIdx.x * 4; c < C; c += blockDim.x * 4) {
        const float4 hv = *reinterpret_cast<const float4*>(hr + c);
        atomicAdd(orow + c + 0, hv.x * w);
        atomicAdd(orow + c + 1, hv.y * w);
        atomicAdd(orow + c + 2, hv.z * w);
        atomicAdd(orow + c + 3, hv.w * w);
    }
}

// grid: (cdiv(C,256), n)
__global__ void bn_relu_kernel(float* __restrict__ x,
                               const float* __restrict__ g,
                               const float* __restrict__ be,
                               const float* __restrict__ m,
                               const float* __restrict__ v, int C) {
    const int node = blockIdx.y;
    const int c    = blockIdx.x * blockDim.x + threadIdx.x;
    if (c < C) {
        const size_t i = (size_t)node * C + c;
        float t = (x[i] - m[c]) * rsqrtf(v[c] + BN_EPS) * g[c] + be[c];
        x[i] = t > 0.0f ? t : 0.0f;
    }
}

// ----------------------------------------------------------------- driver ---
extern "C" void kernel_launch(void* const* d_in, const int* in_sizes, int n_in,
                              void* d_out, int out_size, void* d_ws, size_t ws_size,
                              hipStream_t stream) {
    const float* x   = (const float*)d_in[0];
    const int*   ei  = (const int*)d_in[1];
    const float* W1  = (const float*)d_in[2];
    const float* b1  = (const float*)d_in[3];
    const float* g1  = (const float*)d_in[4];
    const float* be1 = (const float*)d_in[5];
    const float* m1  = (const float*)d_in[6];
    const float* v1  = (const float*)d_in[7];
    const float* W2  = (const float*)d_in[8];
    const float* b2  = (const float*)d_in[9];
    const float* g2  = (const float*)d_in[10];
    const float* be2 = (const float*)d_in[11];
    const float* m2  = (const float*)d_in[12];
    const float* v2  = (const float*)d_in[13];
    const float* W3  = (const float*)d_in[14];
    const float* b3  = (const float*)d_in[15];

    const int IN = 1280, H = 256, OUT = 500;
    const int n = in_sizes[0] / IN;     // 50000
    const int E = in_sizes[1] / 2;      // 1600000
    const int* src = ei;
    const int* dst = ei + E;

    float* dinv = (float*)d_ws;
    float* bufG = dinv + n;                         // n x 512 (GEMM out, ld = N)
    float* bufH = bufG + (size_t)n * 512;           // n x 256 (agg out)
    float* out  = (float*)d_out;

    const dim3 gridH (cdiv(H, 256),  n);
    const dim3 gridO (cdiv(OUT, 256), n);

    // node degrees -> dinv
    deg_init_kernel <<<cdiv(n, 256), 256, 0, stream>>>(dinv, n);
    deg_accum_kernel<<<cdiv(E, 256), 256, 0, stream>>>(dst, E, dinv);
    deg_rsqrt_kernel<<<cdiv(n, 256), 256, 0, stream>>>(dinv, n);

    // ---- layer 1: x[n,1280] @ W1 -> H
    launch_gemm(x, W1, bufG, n, H, IN, stream);
    agg_init_kernel <<<gridH, 256, 0, stream>>>(bufG, dinv, b1, bufH, H);
    agg_edges_kernel<<<E, 64, 0, stream>>>(bufG, dinv, src, dst, bufH, H);
    bn_relu_kernel  <<<gridH, 256, 0, stream>>>(bufH, g1, be1, m1, v1, H);

    // ---- layer 2: [n,256] @ W2 -> H
    launch_gemm(bufH, W2, bufG, n, H, H, stream);
    agg_init_kernel <<<gridH, 256, 0, stream>>>(bufG, dinv, b2, bufH, H);
    agg_edges_kernel<<<E, 64, 0, stream>>>(bufG, dinv, src, dst, bufH, H);
    bn_relu_kernel  <<<gridH, 256, 0, stream>>>(bufH, g2, be2, m2, v2, H);

    // ---- layer 3: [n,256] @ W3 -> OUT (500), result straight into d_out
    launch_gemm(bufH, W3, bufG, n, OUT, H, stream);
    agg_init_kernel <<<gridO, 256, 0, stream>>>(bufG, dinv, b3, out, OUT);
    agg_edges_kernel<<<E, 128, 0, stream>>>(bufG, dinv, src, dst, out, OUT);
}